// VectorQuantizerEMA_12008728560136
// MI455X (gfx1250) — compile-verified
//
#include <hip/hip_runtime.h>
#include <hip/hip_bf16.h>

typedef __attribute__((ext_vector_type(2))) float v2f;
typedef __attribute__((ext_vector_type(8))) float v8f;
typedef __attribute__((ext_vector_type(4))) unsigned int v4u;
typedef __attribute__((ext_vector_type(8))) int v8i;
typedef __attribute__((ext_vector_type(4))) int v4i;

#define VQ_N      32768      // B*H*W = 32*32*32
#define VQ_K      4096       // codebook size
#define VQ_C      256        // channels
#define VQ_BETA   0.25f
#define BSTR      258        // LDS row stride in dwords: 256 + TDM pad of 2 dwords
#define ZQ_ELEMS  8388608    // 32*256*32*32

// ---------------------------------------------------------------------------
// TDM: DMA one 16x256 f32 codebook tile (dense, row-major, stride 256) from
// global into LDS at lds_off, padding each 256-dword row with 2 dwords so the
// LDS image has stride BSTR=258 (bank-conflict-free b64 fragment reads).
// D# fields per CDNA5 ISA. 6-arg builtin (clang-23 /
// therock-10.0 signature): (v4u, v8i, v4i, v4i, v8i, i32 cpol).
// ---------------------------------------------------------------------------
__device__ __forceinline__ void tdm_load_tile(unsigned lds_off, const float* gptr) {
  const unsigned long long ga = (unsigned long long)(size_t)gptr;
  v4u g0;
  g0.x = 1u;                                   // count=1 (valid user descriptor)
  g0.y = lds_off;                              // lds_addr (bytes)
  g0.z = (unsigned)(ga & 0xFFFFFFFFu);         // global_addr[31:0]
  g0.w = (unsigned)((ga >> 32) & 0x1FFFFFFu)   // global_addr[56:32]
         | (2u << 30);                         // type=2 ("image")
  v8i g1;
  g1[0] = (int)((2u << 16)      // data_size = 4 bytes
              | (1u << 20)      // pad_enable
              | (7u << 22)      // pad_interval: 256 dwords (one tile row)
              | (1u << 25));    // pad_amount: 2 dwords -> row stride 258
  g1[1] = (int)(256u << 16);    // tensor_dim0 = 256 (bits 63:48; hi half in g1[2][15:0]=0)
  g1[2] = (int)(16u << 16);     // tensor_dim1 = 16  (bits 95:80; hi half in g1[3][15:0]=0)
  g1[3] = (int)(256u << 16);    // tile_dim0 = 256   (bits 127:112)
  g1[4] = 16;                   // tile_dim1 = 16, tile_dim2 = 0
  g1[5] = 256;                  // tensor_dim0_stride = 256 (bits 191:160)
  g1[6] = 0;                    // stride hi / tensor_dim1_stride lo
  g1[7] = 0;                    // tensor_dim1_stride hi
  const v4i gz4 = {0, 0, 0, 0};           // groups 2/3: <=2-D tensor, unused
  const v8i gz8 = {0, 0, 0, 0, 0, 0, 0, 0};
  __builtin_amdgcn_tensor_load_to_lds(g0, g1, gz4, gz4, gz8, 0);
}

// ---------------------------------------------------------------------------
// Kernel 1: per-code squared norms ||e_k||^2, and zero usage counters.
// ---------------------------------------------------------------------------
__global__ __launch_bounds__(256) void vq_init_kernel(
    const float* __restrict__ embed, float* __restrict__ eNorm,
    int* __restrict__ counts) {
  const int k = blockIdx.x * 256 + threadIdx.x;
  const float4* row = (const float4*)(embed + (size_t)k * VQ_C);
  float s = 0.0f;
#pragma unroll 8
  for (int i = 0; i < VQ_C / 4; ++i) {
    float4 v = row[i];
    s = __builtin_fmaf(v.x, v.x, s);
    s = __builtin_fmaf(v.y, v.y, s);
    s = __builtin_fmaf(v.z, v.z, s);
    s = __builtin_fmaf(v.w, v.w, s);
  }
  eNorm[k] = s;
  counts[k] = 0;
}

// ---------------------------------------------------------------------------
// Kernel 2: fused distance + argmin. A strips in VGPRs (WMMA A-fragment
// layout); B tiles double-buffered in LDS via TDM; 64x v_wmma_f32_16x16x4_f32
// per 16x16 tile per chunk.
// ---------------------------------------------------------------------------
__global__ __launch_bounds__(256, 1) void vq_argmin_kernel(
    const float* __restrict__ z, const float* __restrict__ embed,
    const float* __restrict__ eNorm, int* __restrict__ idxOut) {
  __shared__ float Bs[2][16 * BSTR];

  const int t    = threadIdx.x;
  const int lane = t & 31;
  const int wave = t >> 5;
  const int m    = lane & 15;          // M row of A strip / N col of B
  const int kh   = (lane >> 4) << 1;   // 0 or 2: this lane's K pair

  // Kick off TDM for chunk 0 immediately; overlaps with A-fragment loads.
  if (wave == 0) {
    tdm_load_tile((unsigned)(size_t)&Bs[0][0], embed);
  }

  // z_flat[row, c] = z_e[b, c, h, w] = z[b*262144 + c*1024 + p]
  const int row = blockIdx.x * 128 + wave * 16 + m;
  const int bb  = row >> 10;
  const int p   = row & 1023;
  const float* zrow = z + (size_t)bb * (VQ_C * 1024) + p;

  // Stage A strip into VGPRs, already in WMMA 16x4 A-fragment layout.
  v2f a[64];
#pragma unroll
  for (int i = 0; i < 64; ++i) {
    const int c = i * 4 + kh;
    a[i].x = zrow[(size_t)c * 1024];
    a[i].y = zrow[(size_t)(c + 1) * 1024];
  }

  float best[8];
  int   bidx[8];
#pragma unroll
  for (int v = 0; v < 8; ++v) { best[v] = 3.4e38f; bidx[v] = 0; }

  for (int chunk = 0; chunk < VQ_K / 16; ++chunk) {
    const int buf = chunk & 1;
    if (wave == 0) {
      if (chunk + 1 < VQ_K / 16) {
        // Prefetch next tile into the other buffer (safe: end-of-iter barrier
        // of the previous iteration guarantees no wave still reads it).
        tdm_load_tile((unsigned)(size_t)&Bs[buf ^ 1][0],
                      embed + (size_t)(chunk + 1) * (16 * VQ_C));
        // In-order TENSORcnt: <=1 outstanding means current tile has landed.
        __builtin_amdgcn_s_wait_tensorcnt(1);
      } else {
        __builtin_amdgcn_s_wait_tensorcnt(0);
      }
    }
    __syncthreads();  // current buffer visible to all waves

    v8f acc = (v8f)0.0f;
#pragma unroll
    for (int k = 0; k < 64; ++k) {
      v2f bfrag = *(const v2f*)&Bs[buf][m * BSTR + k * 4 + kh];
      acc = __builtin_amdgcn_wmma_f32_16x16x4_f32(
          false, a[k], false, bfrag, (short)0, acc, false, false);
    }

    // D layout: VGPR v -> M = v + 8*(lane/16), N = lane%16.
    const int nGlob = (chunk << 4) + m;
    const float en  = eNorm[nGlob];
#pragma unroll
    for (int v = 0; v < 8; ++v) {
      const float d = __builtin_fmaf(-2.0f, acc[v], en);
      if (d < best[v]) { best[v] = d; bidx[v] = nGlob; }  // strict <: first occurrence
    }
    __syncthreads();  // all reads of Bs[buf^1] (prev iter) + Bs[buf] done
  }

  // Reduce min/argmin over the N dimension (16 lanes per half-wave).
#pragma unroll
  for (int v = 0; v < 8; ++v) {
    float bv = best[v];
    int   bi = bidx[v];
#pragma unroll
    for (int off = 1; off < 16; off <<= 1) {
      const float ov = __shfl_xor(bv, off, 32);
      const int   oi = __shfl_xor(bi, off, 32);
      if (ov < bv || (ov == bv && oi < bi)) { bv = ov; bi = oi; }  // tie -> smaller idx
    }
    if (m == 0) {
      const int outRow = blockIdx.x * 128 + wave * 16 + ((lane >> 4) << 3) + v;
      idxOut[outRow] = bi;
    }
  }
}

// ---------------------------------------------------------------------------
// Kernel 3: gather z_q, write transposed output, per-block loss partials,
// usage counts.
// ---------------------------------------------------------------------------
__global__ __launch_bounds__(256) void vq_gather_kernel(
    const float* __restrict__ z, const float* __restrict__ embed,
    const int* __restrict__ idx, float* __restrict__ out,
    float* __restrict__ partial, int* __restrict__ counts) {
  const int t = threadIdx.x;
  const int n = blockIdx.x * 256 + t;
  const int code = idx[n];
  atomicAdd(&counts[code], 1);

  const int bb = n >> 10;
  const int p  = n & 1023;
  const float* er   = embed + (size_t)code * VQ_C;
  const float* zr   = z   + (size_t)bb * (VQ_C * 1024) + p;
  float*       orow = out + (size_t)bb * (VQ_C * 1024) + p;

  float s = 0.0f;
#pragma unroll 4
  for (int c = 0; c < VQ_C; ++c) {
    const float q  = er[c];                 // gather (L2-resident 4MB table)
    const float ze = zr[(size_t)c * 1024];  // coalesced across threads
    const float d  = q - ze;
    s = __builtin_fmaf(d, d, s);
    orow[(size_t)c * 1024] = q;             // coalesced; z_q_st == z_q numerically
  }

  __shared__ float red[256];
  red[t] = s;
  __syncthreads();
#pragma unroll
  for (int st = 128; st > 0; st >>= 1) {
    if (t < st) red[t] += red[t + st];
    __syncthreads();
  }
  if (t == 0) partial[blockIdx.x] = red[0];
}

// ---------------------------------------------------------------------------
// Kernel 4: deterministic finalize -> loss and utilization scalars.
// ---------------------------------------------------------------------------
__global__ __launch_bounds__(256) void vq_finalize_kernel(
    const float* __restrict__ partial, const int* __restrict__ counts,
    float* __restrict__ out_tail) {
  __shared__ float fred[256];
  __shared__ int   ired[256];
  const int t = threadIdx.x;
  fred[t] = (t < 128) ? partial[t] : 0.0f;
  int cnt = 0;
#pragma unroll
  for (int i = t; i < VQ_K; i += 256) cnt += (counts[i] > 0) ? 1 : 0;
  ired[t] = cnt;
  __syncthreads();
#pragma unroll
  for (int st = 128; st > 0; st >>= 1) {
    if (t < st) { fred[t] += fred[t + st]; ired[t] += ired[t + st]; }
    __syncthreads();
  }
  if (t == 0) {
    out_tail[0] = fred[0] * (VQ_BETA / ((float)VQ_N * (float)VQ_C));
    out_tail[1] = (float)ired[0] / (float)VQ_K;
  }
}

// ---------------------------------------------------------------------------
extern "C" void kernel_launch(void* const* d_in, const int* in_sizes, int n_in,
                              void* d_out, int out_size, void* d_ws, size_t ws_size,
                              hipStream_t stream) {
  const float* z     = (const float*)d_in[0];   // [32,256,32,32] f32
  const float* embed = (const float*)d_in[1];   // [4096,256] f32
  float* out = (float*)d_out;                   // 8388608 z_q + loss + utilization

  // Workspace layout
  float* eNorm   = (float*)d_ws;                                // 4096 f32
  int*   idx     = (int*)((char*)d_ws + (size_t)VQ_K * 4);      // 32768 i32
  int*   counts  = (int*)((char*)idx + (size_t)VQ_N * 4);       // 4096 i32
  float* partial = (float*)((char*)counts + (size_t)VQ_K * 4);  // 128 f32

  vq_init_kernel<<<VQ_K / 256, 256, 0, stream>>>(embed, eNorm, counts);
  vq_argmin_kernel<<<VQ_N / 128, 256, 0, stream>>>(z, embed, eNorm, idx);
  vq_gather_kernel<<<VQ_N / 256, 256, 0, stream>>>(z, embed, idx, out, partial, counts);
  vq_finalize_kernel<<<1, 256, 0, stream>>>(partial, counts, out + ZQ_ELEMS);
}